// EnergyDipolesMACE_41987600285864
// MI455X (gfx1250) — compile-verified
//
#include <hip/hip_runtime.h>
#include <hip/hip_bf16.h>
#include <math.h>

// ---------------------------------------------------------------------------
// MACE-like energy/dipole/forces model for gfx1250 (MI455X).
// All K=64 GEMM-shaped work (radial MLP fwd/bwd, Wmix/Wg/Wsc fwd/bwd) runs
// through V_WMMA_F32_16X16X4_F32 (f32 matches reference dtype exactly).
// GEMM kernel is templated on <FLAGS, K, NC> so the inner loop is branch-free,
// fully unrolled, and the epilogue has no per-lane column guards for NC=64.
// Workspace assumption: ~850 MB of d_ws.
// ---------------------------------------------------------------------------

#define RMAX  5.0f
#define PI_F  3.14159265358979323846f
#define TPB   256

typedef float v2f __attribute__((ext_vector_type(2)));
typedef float v8f __attribute__((ext_vector_type(8)));

__device__ __forceinline__ float fast_sig(float x) {
  return __builtin_amdgcn_rcpf(1.0f + __expf(-x));   // v_exp_f32 + v_rcp_f32
}
__device__ __forceinline__ float siluf(float x) { return x * fast_sig(x); }
__device__ __forceinline__ float silu_grad(float x) {
  float s = fast_sig(x);
  return s * (1.0f + x * (1.0f - s));
}

enum { GF_TRANSB = 1, GF_ACTIN = 2, GF_DACT = 4, GF_ACCUM = 8 };

// ---------------------------------------------------------------------------
// Generic WMMA GEMM: C[M x NC] = op(A[M x KC]) * B + bias, NC <= 64.
// B staged in LDS as [n][k] (padded stride 68 -> conflict-free ds_load_b64 of
// the B fragment pair; transB handled at stage time; cols >= NC zero-padded
// so NC==8 works). Each wave owns one 16-row stripe. Fragment layouts per
// CDNA5 ISA 7.12.2:
//   A frag v in {0,1}: A[M = lane%16][K = kk + v + 2*(lane/16)]  (one b64 load)
//   B frag v:          B[K = kk + v + 2*(lane/16)][N = n0 + lane%16]
//   C frag v in 0..7:  C[M = row0 + v + 8*(lane/16)][N = n0 + lane%16]
// ---------------------------------------------------------------------------
template <int FLAGS, int KC, int NC>
__global__ __launch_bounds__(256) void k_gemm_wmma(
    const float* __restrict__ A, int lda,
    const float* __restrict__ B, int ldb,
    const float* __restrict__ bias,
    const float* __restrict__ dact,   // pre-activation buffer (same lda as A)
    float* __restrict__ C, int ldc,
    int M)
{
  constexpr int LDSB = 68;                 // padded k-stride (floats)
  constexpr bool TAIL = (NC & 15) != 0;    // partial tile (NC==8 case)
  constexpr int NTILES = (NC + 15) / 16;
  __shared__ float Bl[64 * LDSB];
  for (int idx = threadIdx.x; idx < KC * 64; idx += 256) {
    int k = idx & (KC - 1), n = idx / KC;
    float v = 0.0f;
    if (n < NC) v = (FLAGS & GF_TRANSB) ? B[n * ldb + k] : B[k * ldb + n];
    Bl[n * LDSB + k] = v;
  }
  __syncthreads();

  int wave  = threadIdx.x >> 5;
  int lane  = threadIdx.x & 31;
  int row0  = (blockIdx.x * 8 + wave) * 16;
  if (row0 >= M) return;                   // wave-uniform: EXEC stays all-ones
  int lhalf = lane >> 4;                   // 0 or 1
  int l16   = lane & 15;
  int arow  = row0 + l16;
  const float* Arow = A + (long long)arow * lda;
  const float* Drow = (FLAGS & GF_DACT) ? dact + (long long)arow * lda : nullptr;
  __builtin_prefetch(Arow, 0, 1);          // global_prefetch_b8

#pragma unroll
  for (int t = 0; t < NTILES; ++t) {
    int n0   = t * 16;
    int bcol = n0 + l16;
    bool colok = !TAIL || (bcol < NC);     // compile-time true for NC=64
    const float* Brow = Bl + bcol * LDSB;
    float binit = (bias != nullptr && colok) ? bias[bcol] : 0.0f;
    v8f acc;
#pragma unroll
    for (int v = 0; v < 8; ++v) {
      float cv = binit;
      if (FLAGS & GF_ACCUM) {
        int m = row0 + v + 8 * lhalf;
        if (colok) cv += C[(long long)m * ldc + bcol];
      }
      acc[v] = cv;
    }
#pragma unroll
    for (int kk = 0; kk < KC; kk += 4) {
      int kbase = kk + 2 * lhalf;
      v2f a = *(const v2f*)(Arow + kbase);           // global_load_b64
      if (FLAGS & GF_ACTIN) { a[0] = siluf(a[0]); a[1] = siluf(a[1]); }
      if (FLAGS & GF_DACT) {
        v2f d = *(const v2f*)(Drow + kbase);
        a[0] *= silu_grad(d[0]); a[1] *= silu_grad(d[1]);
      }
      v2f b = *(const v2f*)(Brow + kbase);           // ds_load_b64, no conflicts
      // (neg_a, A, neg_b, B, c_mod, C, reuse_a, reuse_b)
      acc = __builtin_amdgcn_wmma_f32_16x16x4_f32(
          false, a, false, b, (short)0, acc, false, false);
    }
#pragma unroll
    for (int v = 0; v < 8; ++v) {
      int m = row0 + v + 8 * lhalf;
      if (colok) C[(long long)m * ldc + bcol] = acc[v];
    }
  }
}

// ---------------------------------------------------------------------------
__global__ __launch_bounds__(256) void k_zero(float* __restrict__ p, long long n) {
  long long i = (long long)blockIdx.x * blockDim.x + threadIdx.x;
  long long stride = (long long)gridDim.x * blockDim.x;
  for (; i < n; i += stride) p[i] = 0.0f;
}

// node_e0, e0 segment-sum, charge*position dipole baseline
__global__ __launch_bounds__(256) void k_node_scalar(
    const float* __restrict__ attrs, const float* __restrict__ aE,
    const float* __restrict__ charges, const float* __restrict__ pos,
    const int* __restrict__ batch, float* __restrict__ out, int N, int G)
{
  int n = blockIdx.x * blockDim.x + threadIdx.x;
  if (n >= N) return;
  float e = 0.0f;
#pragma unroll
  for (int s = 0; s < 10; ++s) e += attrs[n * 10 + s] * aE[s];
  out[G + n] = e;                          // node_e0 output
  int b = batch[n];
  atomicAdd(&out[b], e);                   // e0 into total_energy
  float q = charges[n];
  float* td = out + G + 7LL * N + (long long)b * 3;
  atomicAdd(&td[0], q * pos[n * 3 + 0]);
  atomicAdd(&td[1], q * pos[n * 3 + 1]);
  atomicAdd(&td[2], q * pos[n * 3 + 2]);
}

// feats0[n][l=0][f] = attrs @ W_embed; l>0 zero. layout [n][l][f], f fastest.
__global__ __launch_bounds__(256) void k_embed(
    const float* __restrict__ attrs, const float* __restrict__ Wemb,
    float* __restrict__ feats0, int N)
{
  int t = blockIdx.x * blockDim.x + threadIdx.x;
  if (t >= N * 64) return;
  int n = t >> 6, f = t & 63;
  float v = 0.0f;
#pragma unroll
  for (int s = 0; s < 10; ++s) v += attrs[n * 10 + s] * Wemb[s * 64 + f];
  float* row = feats0 + (long long)n * 576;
  row[f] = v;
#pragma unroll
  for (int l = 1; l < 9; ++l) row[l * 64 + f] = 0.0f;
}

// per-edge geometry: uhat, r, sh(9), bessel(8), cutoff
__global__ __launch_bounds__(256) void k_edge_geom(
    const float* __restrict__ pos, const float* __restrict__ shifts,
    const int* __restrict__ eidx, float* __restrict__ uhat,
    float* __restrict__ rbuf, float* __restrict__ sh,
    float* __restrict__ bess, float* __restrict__ fc, int E)
{
  int e = blockIdx.x * blockDim.x + threadIdx.x;
  if (e >= E) return;
  int s = eidx[e], r_ = eidx[E + e];
  float vx = pos[r_ * 3 + 0] - pos[s * 3 + 0] + shifts[e * 3 + 0];
  float vy = pos[r_ * 3 + 1] - pos[s * 3 + 1] + shifts[e * 3 + 1];
  float vz = pos[r_ * 3 + 2] - pos[s * 3 + 2] + shifts[e * 3 + 2];
  float r2 = vx * vx + vy * vy + vz * vz + 1e-12f;
  float inv = __builtin_amdgcn_rsqf(r2);   // v_rsq_f32
  float r = r2 * inv;
  float x = vx * inv, y = vy * inv, z = vz * inv;
  uhat[e * 3 + 0] = x; uhat[e * 3 + 1] = y; uhat[e * 3 + 2] = z;
  rbuf[e] = r;
  const float c1 = 1.7320508075688772f;   // sqrt(3)
  const float c2 = 3.8729833462074170f;   // sqrt(15)
  const float c3 = 1.1180339887498949f;   // sqrt(5)/2
  float* she = sh + (long long)e * 9;
  she[0] = 1.0f;
  she[1] = c1 * x; she[2] = c1 * y; she[3] = c1 * z;
  she[4] = c2 * x * y; she[5] = c2 * y * z;
  she[6] = c3 * (3.0f * z * z - 1.0f);
  she[7] = c2 * x * z;
  she[8] = 0.5f * c2 * (x * x - y * y);
  const float C0 = 0.6324555320336759f;   // sqrt(2/5)
#pragma unroll
  for (int n = 1; n <= 8; ++n)
    bess[(long long)e * 8 + n - 1] = C0 * __sinf((float)n * PI_F * r / RMAX) * inv;
  float u = r / RMAX, fcv = 0.0f;
  if (u < 1.0f) {
    float u5 = u * u; u5 *= u5; u5 *= u;  // u^5
    fcv = 1.0f - 21.0f * u5 + 35.0f * u5 * u - 15.0f * u5 * u * u;
  }
  fc[e] = fcv;
}

// per-(edge,f): s = sum_l xs*sh; scatter Rw*s*sh/16 into agg[recv]
__global__ __launch_bounds__(256) void k_message(
    const float* __restrict__ feats_in, const float* __restrict__ rwraw,
    const float* __restrict__ fc, const float* __restrict__ sh,
    const int* __restrict__ eidx, float* __restrict__ agg, int E)
{
  long long t = (long long)blockIdx.x * blockDim.x + threadIdx.x;
  if (t >= (long long)E * 64) return;
  int e = (int)(t >> 6), f = (int)(t & 63);
  int snd = eidx[e], rcv = eidx[E + e];
  const float* she = sh + (long long)e * 9;
  const float* xr = feats_in + (long long)snd * 576 + f;
  float s = 0.0f;
#pragma unroll
  for (int l = 0; l < 9; ++l) s += xr[l * 64] * she[l];
  float base = rwraw[(long long)e * 64 + f] * fc[e] * s * (1.0f / 16.0f);
  float* ar = agg + (long long)rcv * 576 + f;
#pragma unroll
  for (int l = 0; l < 9; ++l) atomicAdd(&ar[l * 64], base * she[l]);
}

// out = mix * silu(gate_pre) broadcast over l
__global__ __launch_bounds__(256) void k_gate_combine(
    const float* __restrict__ mix, const float* __restrict__ gp,
    float* __restrict__ outp, int N)
{
  int t = blockIdx.x * blockDim.x + threadIdx.x;
  if (t >= N * 64) return;
  int n = t >> 6, g = t & 63;
  float gt = siluf(gp[n * 64 + g]);
  const float* mr = mix + (long long)n * 576 + g;
  float* orow = outp + (long long)n * 576 + g;
#pragma unroll
  for (int l = 0; l < 9; ++l) orow[l * 64] = mr[l * 64] * gt;
}

// per-layer node energy + dipole contributions
__global__ __launch_bounds__(256) void k_node_out(
    const float* __restrict__ outp, const float* __restrict__ w_e,
    const float* __restrict__ w_d, const int* __restrict__ batch,
    float* __restrict__ out, int N, int G)
{
  int n = blockIdx.x * blockDim.x + threadIdx.x;
  if (n >= N) return;
  const float* row = outp + (long long)n * 576;
  float e = 0.0f;
  for (int f = 0; f < 64; ++f) e += row[f] * w_e[f];
  int b = batch[n];
  atomicAdd(&out[b], e);
  float* dip = out + G + 4LL * N + (long long)n * 3;
  float* td  = out + G + 7LL * N + (long long)b * 3;
#pragma unroll
  for (int k = 0; k < 3; ++k) {
    float d = 0.0f;
    for (int f = 0; f < 64; ++f) d += row[(1 + k) * 64 + f] * w_d[f];
    dip[k] += d;                 // only writer, sequential layers
    atomicAdd(&td[k], d);
  }
}

__global__ __launch_bounds__(256) void k_gout_init(
    float* __restrict__ g_out, const float* __restrict__ w_e, long long total)
{
  long long t = (long long)blockIdx.x * blockDim.x + threadIdx.x;
  if (t >= total) return;
  int g = (int)(t & 63);
  int l = (int)((t >> 6) % 9);
  g_out[t] = (l == 0) ? w_e[g] : 0.0f;
}

__global__ __launch_bounds__(256) void k_we_add(
    float* __restrict__ g_feats, const float* __restrict__ w_e, int N)
{
  int t = blockIdx.x * blockDim.x + threadIdx.x;
  if (t >= N * 64) return;
  int n = t >> 6, g = t & 63;
  g_feats[(long long)n * 576 + g] += w_e[g];
}

// gate backward: g_mix = g_out*gate; g_gatepre = (sum_l g_out*mix)*silu'(pre)
__global__ __launch_bounds__(256) void k_node_bwd_a(
    const float* __restrict__ g_out, const float* __restrict__ mix,
    const float* __restrict__ gp, float* __restrict__ g_mix,
    float* __restrict__ g_gp, int N)
{
  int t = blockIdx.x * blockDim.x + threadIdx.x;
  if (t >= N * 64) return;
  int n = t >> 6, g = t & 63;
  float pre = gp[n * 64 + g];
  float gt = siluf(pre);
  const float* go = g_out + (long long)n * 576 + g;
  const float* mr = mix + (long long)n * 576 + g;
  float* gm = g_mix + (long long)n * 576 + g;
  float gg = 0.0f;
#pragma unroll
  for (int l = 0; l < 9; ++l) {
    float gol = go[l * 64];
    gg += gol * mr[l * 64];
    gm[l * 64] = gol * gt;
  }
  g_gp[n * 64 + g] = gg * silu_grad(pre);
}

// message VJP: g_Rwraw, g_sh, g_r (cutoff path), optional g_feats[sender]
__global__ __launch_bounds__(256) void k_edge_bwd(
    const float* __restrict__ feats_in, const float* __restrict__ g_agg,
    const float* __restrict__ rwraw, const float* __restrict__ fc,
    const float* __restrict__ sh, const float* __restrict__ rbuf,
    const int* __restrict__ eidx, float* __restrict__ g_rwraw,
    float* __restrict__ g_shA, float* __restrict__ g_rA,
    float* __restrict__ g_feats, int E, int withXs)
{
  long long t = (long long)blockIdx.x * blockDim.x + threadIdx.x;
  if (t >= (long long)E * 64) return;
  int e = (int)(t >> 6), f = (int)(t & 63);
  int snd = eidx[e], rcv = eidx[E + e];
  const float* she = sh + (long long)e * 9;
  const float* ga = g_agg + (long long)rcv * 576 + f;
  const float* xr = feats_in + (long long)snd * 576 + f;
  float gm[9], xs[9], tsum = 0.0f, ssum = 0.0f;
#pragma unroll
  for (int l = 0; l < 9; ++l) {
    gm[l] = ga[l * 64] * (1.0f / 16.0f);
    xs[l] = xr[l * 64];
    tsum += gm[l] * she[l];
    ssum += xs[l] * she[l];
  }
  float Rwr = rwraw[(long long)e * 64 + f];
  float fcv = fc[e];
  float Rw  = Rwr * fcv;
  float gRw = tsum * ssum;
  float gS  = Rw * tsum;
  g_rwraw[(long long)e * 64 + f] = gRw * fcv;
  float r = rbuf[e], u = r / RMAX, dfc = 0.0f;
  if (u < 1.0f) {
    float u4 = u * u; u4 *= u4;
    dfc = (-105.0f * u4 + 210.0f * u4 * u - 105.0f * u4 * u * u) / RMAX;
  }
  atomicAdd(&g_rA[e], gRw * Rwr * dfc);
#pragma unroll
  for (int l = 0; l < 9; ++l)
    atomicAdd(&g_shA[(long long)e * 9 + l], gm[l] * Rw * ssum + gS * xs[l]);
  if (withXs) {
    float* gf = g_feats + (long long)snd * 576 + f;
#pragma unroll
    for (int l = 0; l < 9; ++l) atomicAdd(&gf[l * 64], gS * she[l]);
  }
}

// bessel backward: g_r += sum_n g_bess * d bess_n / dr
__global__ __launch_bounds__(256) void k_bess_to_r(
    const float* __restrict__ g_bess, const float* __restrict__ rbuf,
    float* __restrict__ g_r, int E)
{
  int e = blockIdx.x * blockDim.x + threadIdx.x;
  if (e >= E) return;
  float r = rbuf[e];
  float inv = __builtin_amdgcn_rcpf(r);
  const float C0 = 0.6324555320336759f;
  float acc = 0.0f;
#pragma unroll
  for (int n = 1; n <= 8; ++n) {
    float w = (float)n * PI_F / RMAX;
    float s = __sinf(w * r), c = __cosf(w * r);
    acc += g_bess[(long long)e * 8 + n - 1] * C0 * (w * c * inv - s * inv * inv);
  }
  g_r[e] += acc;
}

// sh Jacobian + r path -> g_vec; scatter to forces (forces = -grad)
__global__ __launch_bounds__(256) void k_force_scatter(
    const float* __restrict__ g_shA, const float* __restrict__ g_rA,
    const float* __restrict__ uhat, const float* __restrict__ rbuf,
    const int* __restrict__ eidx, float* __restrict__ F, int E)
{
  int e = blockIdx.x * blockDim.x + threadIdx.x;
  if (e >= E) return;
  const float* g = g_shA + (long long)e * 9;
  float x = uhat[e * 3 + 0], y = uhat[e * 3 + 1], z = uhat[e * 3 + 2];
  const float c1 = 1.7320508075688772f;
  const float c2 = 3.8729833462074170f;
  const float c3 = 1.1180339887498949f;
  float gx = c1 * g[1] + c2 * (y * g[4] + z * g[7] + x * g[8]);
  float gy = c1 * g[2] + c2 * (x * g[4] + z * g[5] - y * g[8]);
  float gz = c1 * g[3] + c2 * (y * g[5] + x * g[7]) + 6.0f * c3 * z * g[6];
  float gd = gx * x + gy * y + gz * z;
  float inv = __builtin_amdgcn_rcpf(rbuf[e]);
  float gr = g_rA[e];
  float vx = (gx - gd * x) * inv + gr * x;
  float vy = (gy - gd * y) * inv + gr * y;
  float vz = (gz - gd * z) * inv + gr * z;
  int snd = eidx[e], rcv = eidx[E + e];
  atomicAdd(&F[rcv * 3 + 0], -vx); atomicAdd(&F[snd * 3 + 0], vx);
  atomicAdd(&F[rcv * 3 + 1], -vy); atomicAdd(&F[snd * 3 + 1], vy);
  atomicAdd(&F[rcv * 3 + 2], -vz); atomicAdd(&F[snd * 3 + 2], vz);
}

// ---------------------------------------------------------------------------
extern "C" void kernel_launch(void* const* d_in, const int* in_sizes, int n_in,
                              void* d_out, int out_size, void* d_ws, size_t ws_size,
                              hipStream_t stream)
{
  const float* positions  = (const float*)d_in[0];
  const float* node_attrs = (const float*)d_in[1];
  const float* charges    = (const float*)d_in[2];
  const float* shifts     = (const float*)d_in[3];
  // params flattened (jax pytree, sorted keys): W_embed, atomic_energies, layers
  const float* W_embed  = (const float*)d_in[4];
  const float* atomic_E = (const float*)d_in[5];
  const float *Wg[2], *Wmix[2], *Wsc[2], *rW1[2], *rW2[2], *rW3[2],
              *rb1[2], *rb2[2], *w_d[2], *w_e[2];
  for (int i = 0; i < 2; ++i) {   // per-layer keys: Wg,Wmix,Wsc,rW1,rW2,rW3,rb1,rb2,w_d,w_e
    int b = 6 + i * 10;
    Wg[i]  = (const float*)d_in[b + 0]; Wmix[i] = (const float*)d_in[b + 1];
    Wsc[i] = (const float*)d_in[b + 2]; rW1[i]  = (const float*)d_in[b + 3];
    rW2[i] = (const float*)d_in[b + 4]; rW3[i]  = (const float*)d_in[b + 5];
    rb1[i] = (const float*)d_in[b + 6]; rb2[i]  = (const float*)d_in[b + 7];
    w_d[i] = (const float*)d_in[b + 8]; w_e[i]  = (const float*)d_in[b + 9];
  }
  const int* eidx  = (const int*)d_in[26];
  const int* batch = (const int*)d_in[27];

  const int N = in_sizes[2];
  const int E = in_sizes[3] / 3;
  const int G = (out_size - 7 * N) / 4;   // out = [G | N | 3N | 3N | 3G]
  float* out = (float*)d_out;
  float* ws  = (float*)d_ws;
  (void)n_in; (void)ws_size;

  long long o = 0;
  auto alloc = [&](long long n) { float* p = ws + o; o += n; return p; };
  float* uhat   = alloc(3LL * E);  float* rbuf = alloc(E);
  float* shb    = alloc(9LL * E);  float* bess = alloc(8LL * E);
  float* fcb    = alloc(E);
  float* feats0 = alloc(576LL * N);
  float* feats1 = alloc(576LL * N);
  float* mixb0  = alloc(576LL * N);
  float* mixb1  = alloc(576LL * N);
  float* gpb0   = alloc(64LL * N);
  float* gpb1   = alloc(64LL * N);
  float* agg    = alloc(576LL * N);       // also reused as out1 storage
  float* a1     = alloc(64LL * E);
  float* a2     = alloc(64LL * E);
  float* rwraw  = alloc(64LL * E);
  float* g_out  = alloc(576LL * N);
  float* g_mix  = alloc(576LL * N);
  float* g_agg  = alloc(576LL * N);
  float* g_feats= alloc(576LL * N);
  float* g_gp   = alloc(64LL * N);
  float* g_rw   = alloc(64LL * E);
  float* g_h    = alloc(64LL * E);
  float* g_bess = alloc(8LL * E);
  float* g_shA  = alloc(9LL * E);
  float* g_rA   = alloc(E);

  float* mixb[2] = { mixb0, mixb1 };
  float* gpb[2]  = { gpb0, gpb1 };

  auto blocks = [](long long n) { return (int)((n + TPB - 1) / TPB); };

  // compile-time-dispatched GEMM
  auto gemm = [&](const float* A, int lda, const float* B, int ldb,
                  const float* bias, const float* dact,
                  float* C, int ldc, int M, int K, int Ncols, int flags) {
    int grid = (M + 127) / 128;
#define GEMM_GO(F, KC, NC) \
    k_gemm_wmma<F, KC, NC><<<grid, 256, 0, stream>>>(A, lda, B, ldb, bias, dact, C, ldc, M)
    if (K == 8)     { GEMM_GO(0, 8, 64); return; }
    if (Ncols == 8) { GEMM_GO(GF_TRANSB | GF_DACT, 64, 8); return; }
    switch (flags) {
      case 0:                       GEMM_GO(0, 64, 64); break;
      case GF_ACTIN:                GEMM_GO(GF_ACTIN, 64, 64); break;
      case GF_ACCUM:                GEMM_GO(GF_ACCUM, 64, 64); break;
      case GF_TRANSB:               GEMM_GO(GF_TRANSB, 64, 64); break;
      case GF_TRANSB | GF_ACCUM:    GEMM_GO(GF_TRANSB | GF_ACCUM, 64, 64); break;
      case GF_TRANSB | GF_DACT:     GEMM_GO(GF_TRANSB | GF_DACT, 64, 64); break;
      default:                      GEMM_GO(0, 64, 64); break;
    }
#undef GEMM_GO
  };
  auto radial_fwd = [&](int l) {
    gemm(bess, 8,  rW1[l], 64, rb1[l],  nullptr, a1,    64, E, 8,  64, 0);
    gemm(a1,  64,  rW2[l], 64, rb2[l],  nullptr, a2,    64, E, 64, 64, GF_ACTIN);
    gemm(a2,  64,  rW3[l], 64, nullptr, nullptr, rwraw, 64, E, 64, 64, GF_ACTIN);
  };

  // ---------------- forward ----------------
  k_zero<<<blocks(out_size), TPB, 0, stream>>>(out, out_size);
  k_node_scalar<<<blocks(N), TPB, 0, stream>>>(node_attrs, atomic_E, charges,
                                               positions, batch, out, N, G);
  k_embed<<<blocks((long long)N * 64), TPB, 0, stream>>>(node_attrs, W_embed, feats0, N);
  k_edge_geom<<<blocks(E), TPB, 0, stream>>>(positions, shifts, eidx,
                                             uhat, rbuf, shb, bess, fcb, E);
  for (int l = 0; l < 2; ++l) {
    float* fin  = (l == 0) ? feats0 : feats1;
    float* outp = (l == 0) ? feats1 : agg;   // agg dead after mix GEMM -> reuse
    radial_fwd(l);
    k_zero<<<blocks(576LL * N), TPB, 0, stream>>>(agg, 576LL * N);
    k_message<<<blocks((long long)E * 64), TPB, 0, stream>>>(fin, rwraw, fcb, shb, eidx, agg, E);
    gemm(agg,      64,  Wmix[l], 64, nullptr, nullptr, mixb[l], 64,  N * 9, 64, 64, 0);
    gemm(mixb[l], 576,  Wg[l],   64, nullptr, nullptr, gpb[l],  64,  N,     64, 64, 0);
    k_gate_combine<<<blocks((long long)N * 64), TPB, 0, stream>>>(mixb[l], gpb[l], outp, N);
    if (l > 0)
      gemm(fin, 64, Wsc[l], 64, nullptr, nullptr, outp, 64, N * 9, 64, 64, GF_ACCUM);
    k_node_out<<<blocks(N), TPB, 0, stream>>>(outp, w_e[l], w_d[l], batch, out, N, G);
  }

  // ---------------- backward (forces) ----------------
  k_zero<<<blocks(576LL * N), TPB, 0, stream>>>(g_feats, 576LL * N);
  k_zero<<<blocks(9LL * E), TPB, 0, stream>>>(g_shA, 9LL * E);
  k_zero<<<blocks(E), TPB, 0, stream>>>(g_rA, E);

  for (int l = 1; l >= 0; --l) {
    float* fin = (l == 0) ? feats0 : feats1;
    radial_fwd(l);                        // recompute a1,a2,rwraw for layer l
    float* gout;
    if (l == 1) {
      k_gout_init<<<blocks(576LL * N), TPB, 0, stream>>>(g_out, w_e[1], 576LL * N);
      gout = g_out;
    } else {
      k_we_add<<<blocks((long long)N * 64), TPB, 0, stream>>>(g_feats, w_e[0], N);
      gout = g_feats;                     // accumulated from layer 1 + w_e0
    }
    k_node_bwd_a<<<blocks((long long)N * 64), TPB, 0, stream>>>(gout, mixb[l], gpb[l],
                                                                g_mix, g_gp, N);
    gemm(g_gp,  64, Wg[l],   64, nullptr, nullptr, g_mix, 576, N,     64, 64, GF_TRANSB | GF_ACCUM);
    gemm(g_mix, 64, Wmix[l], 64, nullptr, nullptr, g_agg, 64,  N * 9, 64, 64, GF_TRANSB);
    if (l == 1)
      gemm(gout, 64, Wsc[1], 64, nullptr, nullptr, g_feats, 64, N * 9, 64, 64, GF_TRANSB | GF_ACCUM);
    k_edge_bwd<<<blocks((long long)E * 64), TPB, 0, stream>>>(
        fin, g_agg, rwraw, fcb, shb, rbuf, eidx,
        g_rw, g_shA, g_rA, g_feats, E, (l == 1) ? 1 : 0);
    gemm(g_rw, 64, rW3[l], 64, nullptr, nullptr, g_h,    64, E, 64, 64, GF_TRANSB);
    gemm(g_h,  64, rW2[l], 64, nullptr, a2,      g_rw,   64, E, 64, 64, GF_TRANSB | GF_DACT);
    gemm(g_rw, 64, rW1[l], 64, nullptr, a1,      g_bess, 8,  E, 64, 8,  GF_TRANSB | GF_DACT);
    k_bess_to_r<<<blocks(E), TPB, 0, stream>>>(g_bess, rbuf, g_rA, E);
  }
  k_force_scatter<<<blocks(E), TPB, 0, stream>>>(g_shA, g_rA, uhat, rbuf, eidx,
                                                 out + (G + (long long)N), E);
}